// CNNDST_20796231647886
// MI455X (gfx1250) — compile-verified
//
#include <hip/hip_runtime.h>
#include <stdint.h>

// out[b][r] = sum over (I,J) in [0,1024)^2 with (I & J) == r of M[b][I][J]
// (exact closed form of the reference's 10-level quadtree + bit-reversal)

#define LANES 32
#define WAVES 4
#define BLOCK (LANES * WAVES)
#define ROWS_PER_WAVE 8
#define ROWS_PER_WG (WAVES * ROWS_PER_WAVE) /* 32 */
#define IMGW 1024
#define ROWBLOCKS (IMGW / ROWS_PER_WG) /* 32 */

// ---------- CDNA5 async global -> LDS path ----------
#if defined(__has_builtin)
#if __has_builtin(__builtin_amdgcn_global_load_async_to_lds_b128)
#define HAVE_ASYNC_BUILTIN 1
#endif
#if __has_builtin(__builtin_amdgcn_s_wait_asynccnt)
#define HAVE_WAIT_BUILTIN 1
#endif
#endif

typedef int v4i_t __attribute__((vector_size(16)));
typedef __attribute__((address_space(1))) v4i_t gv4i_t;  // global int4
typedef __attribute__((address_space(3))) v4i_t lv4i_t;  // LDS int4

__device__ __forceinline__ void async_copy16(void* lds_dst, const void* gsrc) {
#if defined(HAVE_ASYNC_BUILTIN)
  // (global src, LDS dst, imm offset, imm cpol)
  __builtin_amdgcn_global_load_async_to_lds_b128(
      (gv4i_t*)gsrc, (lv4i_t*)lds_dst, 0, 0);
#else
  unsigned lds_off = (unsigned)(uintptr_t)(lv4i_t*)lds_dst;
  asm volatile("global_load_async_to_lds_b128 %0, %1, off"
               :: "v"(lds_off), "v"(gsrc) : "memory");
#endif
}

#if defined(HAVE_WAIT_BUILTIN)
#define WAIT_ASYNC(n) __builtin_amdgcn_s_wait_asynccnt(n)
#else
#define WAIT_ASYNC(n) asm volatile("s_wait_asynccnt %0" :: "i"(n) : "memory")
#endif

__global__ void CNNDST_zero_kernel(float* __restrict__ out, int n) {
  int i = blockIdx.x * blockDim.x + threadIdx.x;
  if (i < n) out[i] = 0.0f;
}

__global__ __launch_bounds__(BLOCK)
void CNNDST_bins_kernel(const float* __restrict__ M, float* __restrict__ out) {
  __shared__ float bins[IMGW];                 // 4 KB  block-private bins
  __shared__ float stage[WAVES][2][IMGW];      // 32 KB double-buffered rows

  const int tid  = threadIdx.x;
  const int wave = tid >> 5;
  const int lane = tid & 31;
  const int b    = blockIdx.y;
  const int rowBase = blockIdx.x * ROWS_PER_WG + wave * ROWS_PER_WAVE;
  const float* img = M + (size_t)b * IMGW * IMGW;

  for (int i = tid; i < IMGW; i += BLOCK) bins[i] = 0.0f;
  __syncthreads();

  // prime the pipeline: async-load row 0 of this wave (8 x b128 = 4 KB)
  {
    const float* src = img + (size_t)rowBase * IMGW;
    float* dst = &stage[wave][0][0];
#pragma unroll
    for (int i = 0; i < 8; ++i)
      async_copy16(dst + i * 128 + lane * 4, src + i * 128 + lane * 4);
  }

  for (int rr = 0; rr < ROWS_PER_WAVE; ++rr) {
    if (rr + 1 < ROWS_PER_WAVE) {
      const float* src = img + (size_t)(rowBase + rr + 1) * IMGW;
      float* dst = &stage[wave][(rr + 1) & 1][0];
#pragma unroll
      for (int i = 0; i < 8; ++i)
        async_copy16(dst + i * 128 + lane * 4, src + i * 128 + lane * 4);
      WAIT_ASYNC(8);  // async loads complete in order -> row rr is resident
    } else {
      WAIT_ASYNC(0);
    }
    asm volatile("" ::: "memory");

    const int I    = rowBase + rr;
    const int Ilo  = I & 31;
    const int Ihi  = (I >> 5) & 31;
    const int nIhi = (~Ihi) & 31;
    const bool rep = ((lane & ~Ilo) & 31) == 0;  // lane == (lane & Ilo)
    const float* buf = &stage[wave][rr & 1][0];

    // enumerate h = submasks of Ihi: all chunks jhi with (Ihi & jhi) == h
    int h = Ihi;
    while (true) {
      float acc = 0.0f;
      int f = nIhi;  // jhi = h | f, f over submasks of ~Ihi
      while (true) {
        acc += buf[((h | f) << 5) + lane];
        if (f == 0) break;
        f = (f - 1) & nIhi;
      }
      // xor-butterfly over the zero bits of Ilo: groups lanes by (lane & Ilo)
      if (!(Ilo & 1))  acc += __shfl_xor(acc, 1, 32);
      if (!(Ilo & 2))  acc += __shfl_xor(acc, 2, 32);
      if (!(Ilo & 4))  acc += __shfl_xor(acc, 4, 32);
      if (!(Ilo & 8))  acc += __shfl_xor(acc, 8, 32);
      if (!(Ilo & 16)) acc += __shfl_xor(acc, 16, 32);
      // one representative lane per group -> conflict-free LDS f32 atomic
      if (rep) atomicAdd(&bins[(h << 5) | lane], acc);
      if (h == 0) break;
      h = (h - 1) & Ihi;
    }
  }

  __syncthreads();
  for (int r = tid; r < IMGW; r += BLOCK)
    unsafeAtomicAdd(&out[(size_t)b * IMGW + r], bins[r]);
}

extern "C" void kernel_launch(void* const* d_in, const int* in_sizes, int n_in,
                              void* d_out, int out_size, void* d_ws, size_t ws_size,
                              hipStream_t stream) {
  const float* M = (const float*)d_in[0];
  float* out = (float*)d_out;
  const int B = in_sizes[0] >> 20;  // elements / 1024^2

  CNNDST_zero_kernel<<<(out_size + 255) / 256, 256, 0, stream>>>(out, out_size);

  dim3 grid(ROWBLOCKS, B);
  CNNDST_bins_kernel<<<grid, BLOCK, 0, stream>>>(M, out);
}